// GTCNN_83854941487394
// MI455X (gfx1250) — compile-verified
//
#include <hip/hip_runtime.h>
#include <hip/hip_bf16.h>

// Problem constants (from reference)
#define TB 2        // batch
#define TT 64       // temporal nodes
#define TN 128      // spatial nodes
#define TM (TT*TN)  // 8192 product-graph nodes
#define HID 32
#define FIN 32
#define FOUT 16
#define NLAYERS 2
#define KORD 4

typedef float v2f __attribute__((ext_vector_type(2)));
typedef float v8f __attribute__((ext_vector_type(8)));

// D = A(16x4,f32) * B(4x16,f32) + C(16x16,f32)  -> v_wmma_f32_16x16x4_f32
__device__ __forceinline__ v8f wmma_k4(v2f a, v2f b, v8f c) {
    return __builtin_amdgcn_wmma_f32_16x16x4_f32(
        /*neg_a=*/false, a, /*neg_b=*/false, b,
        /*c_mod=*/(short)0, c, /*reuse_a=*/false, /*reuse_b=*/false);
}

// ---------------------------------------------------------------------------
// Kernel: Heff[l] = sum_k H[l,k]  (collapse einsum-k;  [L,32,32])
// ---------------------------------------------------------------------------
__global__ void heff_kernel(const float* __restrict__ H, float* __restrict__ Heff) {
    int i = blockIdx.x * 256 + threadIdx.x;            // 0 .. L*HID*HID-1 = 2047
    if (i < NLAYERS * HID * HID) {
        int l = i / (HID * HID);
        int hf = i % (HID * HID);
        float acc = 0.f;
        for (int k = 0; k < KORD; ++k)
            acc += H[(size_t)(l * KORD + k) * HID * HID + hf];
        Heff[i] = acc;
    }
}

// ---------------------------------------------------------------------------
// Kernel: h = x @ W1 + b1      x:[B*M,32]  W1:[32,32]  h:[B*M,32]
// one wave -> 16 rows x 32 cols (two 16x16 tiles), K=32 in steps of 4
// ---------------------------------------------------------------------------
__global__ __launch_bounds__(256) void inproj_kernel(
    const float* __restrict__ x, const float* __restrict__ W1,
    const float* __restrict__ b1, float* __restrict__ h)
{
    int gwave = (blockIdx.x * blockDim.x + threadIdx.x) >> 5;   // 0..1023
    int lane  = threadIdx.x & 31;
    int lo = lane & 15, half = lane >> 4;
    int rowBase = gwave * 16;

    const float* xr = x + (size_t)(rowBase + lo) * FIN;
    v8f c0 = {0,0,0,0,0,0,0,0}, c1 = {0,0,0,0,0,0,0,0};
    for (int k0 = 0; k0 < FIN; k0 += 4) {
        int ka = k0 + 2 * half;
        v2f a = *(const v2f*)(xr + ka);
        v2f bA; bA.x = W1[(size_t)ka * HID + lo];        bA.y = W1[(size_t)(ka + 1) * HID + lo];
        v2f bB; bB.x = W1[(size_t)ka * HID + lo + 16];   bB.y = W1[(size_t)(ka + 1) * HID + lo + 16];
        c0 = wmma_k4(a, bA, c0);
        c1 = wmma_k4(a, bB, c1);
    }
    for (int j = 0; j < 8; ++j) {
        int r = rowBase + j + 8 * half;
        h[(size_t)r * HID + lo]      = c0[j] + b1[lo];
        h[(size_t)r * HID + lo + 16] = c1[j] + b1[lo + 16];
    }
}

// ---------------------------------------------------------------------------
// Kernel (spatial): per (b,t) block:  g = Adj_s(128x128) @ h[b,t](128x32)
//   write u = s0*h + s1*g   and   v = s2*h + s3*g
// 512 threads = 16 waves; wave w -> rowTile=w/2 (8), colTile=w%2 (2)
// ---------------------------------------------------------------------------
__global__ __launch_bounds__(512) void spatial_kernel(
    const float* __restrict__ Adj_s,   // [128,128]
    const float* __restrict__ h,       // [B,T,N,32]
    const float* __restrict__ s,       // [4]
    float* __restrict__ ubuf, float* __restrict__ vbuf)
{
    __shared__ float hsT[HID][TN + 2];     // transposed h tile, stride 130 -> conflict free
    int bt = blockIdx.x;                   // 0 .. B*T-1
    const float* hb = h + (size_t)bt * TN * HID;

    for (int i = threadIdx.x; i < TN * HID; i += blockDim.x) {
        int n = i >> 5, f = i & 31;
        hsT[f][n] = hb[i];
    }
    __syncthreads();

    int wave = threadIdx.x >> 5, lane = threadIdx.x & 31;
    int lo = lane & 15, half = lane >> 4;
    int rowTile = wave >> 1, colTile = wave & 1;
    int row = rowTile * 16 + lo;           // output spatial node (A-matrix row)
    int col = colTile * 16 + lo;           // hidden index

    v8f c = {0,0,0,0,0,0,0,0};
    const float* arow = Adj_s + (size_t)row * TN;
    for (int k0 = 0; k0 < TN; k0 += 4) {
        int ka = k0 + 2 * half;
        v2f a = *(const v2f*)(arow + ka);
        v2f b; b.x = hsT[col][ka]; b.y = hsT[col][ka + 1];  // wait: B[k][col] = h[k][col] = hsT[col][k]
        c = wmma_k4(a, b, c);
    }

    float s0 = s[0], s1 = s[1], s2 = s[2], s3 = s[3];
    float* ub = ubuf + (size_t)bt * TN * HID;
    float* vb = vbuf + (size_t)bt * TN * HID;
    for (int j = 0; j < 8; ++j) {
        int r = rowTile * 16 + j + 8 * half;
        float g  = c[j];
        float hv = hsT[col][r];
        ub[(size_t)r * HID + col] = s0 * hv + s1 * g;
        vb[(size_t)r * HID + col] = s2 * hv + s3 * g;
    }
}

// ---------------------------------------------------------------------------
// Kernel (temporal + mix + tanh): per (b,n) block:
//   w  = Adj_t(64x64) @ v[b,:,n,:](64x32)
//   Ah = u + w
//   hnext = tanh(Ah @ Heff[l])
// 256 threads = 8 waves; wave w -> rowTile=w/2 (4 over T), colTile=w%2
// ---------------------------------------------------------------------------
__global__ __launch_bounds__(256) void temporal_kernel(
    const float* __restrict__ Adj_t,   // [64,64]
    const float* __restrict__ ubuf,    // [B,T,N,32]
    const float* __restrict__ vbuf,    // [B,T,N,32]
    const float* __restrict__ Heff,    // [32,32] (this layer)
    float* __restrict__ hnext)         // [B,T,N,32]
{
    __shared__ float vt[HID][TT + 2];  // v transposed: vt[f][t]
    __shared__ float ah[HID][TT + 2];  // Ah transposed: ah[f][t]
    int bn = blockIdx.x;               // 0 .. B*N-1
    int b = bn / TN, n = bn % TN;
    const float* vb = vbuf + ((size_t)b * TT * TN + n) * HID;
    const float* ub = ubuf + ((size_t)b * TT * TN + n) * HID;
    float*       hn = hnext + ((size_t)b * TT * TN + n) * HID;

    for (int i = threadIdx.x; i < TT * HID; i += blockDim.x) {
        int t = i >> 5, f = i & 31;
        vt[f][t] = vb[(size_t)t * TN * HID + f];
    }
    __syncthreads();

    int wave = threadIdx.x >> 5, lane = threadIdx.x & 31;
    int lo = lane & 15, half = lane >> 4;
    int rowTile = wave >> 1, colTile = wave & 1;
    int row = rowTile * 16 + lo;       // t index (A-matrix row)
    int col = colTile * 16 + lo;       // hidden index

    // --- temporal matmul: K over t_in (64) ---
    v8f c = {0,0,0,0,0,0,0,0};
    const float* arow = Adj_t + (size_t)row * TT;
    for (int k0 = 0; k0 < TT; k0 += 4) {
        int ka = k0 + 2 * half;
        v2f a = *(const v2f*)(arow + ka);
        v2f bv; bv.x = vt[col][ka]; bv.y = vt[col][ka + 1];
        c = wmma_k4(a, bv, c);
    }
    // Ah = u + w  -> LDS (transposed)
    for (int j = 0; j < 8; ++j) {
        int t = rowTile * 16 + j + 8 * half;
        ah[col][t] = ub[(size_t)t * TN * HID + col] + c[j];
    }
    __syncthreads();

    // --- mixing matmul: hnext[t,f] = tanh(sum_h Ah[t,h] * Heff[h,f]), K=32 ---
    v8f c2 = {0,0,0,0,0,0,0,0};
    for (int k0 = 0; k0 < HID; k0 += 4) {
        int ka = k0 + 2 * half;
        v2f a; a.x = ah[ka][row]; a.y = ah[ka + 1][row];
        v2f bv; bv.x = Heff[(size_t)ka * HID + col]; bv.y = Heff[(size_t)(ka + 1) * HID + col];
        c2 = wmma_k4(a, bv, c2);
    }
    for (int j = 0; j < 8; ++j) {
        int t = rowTile * 16 + j + 8 * half;
        hn[(size_t)t * TN * HID + col] = tanhf(c2[j]);
    }
}

// ---------------------------------------------------------------------------
// Kernel: out = h @ W2 + b2     h:[B*M,32]  W2:[32,16]  out:[B*M,16]
// one wave -> 16 rows x 16 cols, K=32
// ---------------------------------------------------------------------------
__global__ __launch_bounds__(256) void outproj_kernel(
    const float* __restrict__ h, const float* __restrict__ W2,
    const float* __restrict__ b2, float* __restrict__ out)
{
    int gwave = (blockIdx.x * blockDim.x + threadIdx.x) >> 5;   // 0..1023
    int lane  = threadIdx.x & 31;
    int lo = lane & 15, half = lane >> 4;
    int rowBase = gwave * 16;

    const float* hr = h + (size_t)(rowBase + lo) * HID;
    v8f c = {0,0,0,0,0,0,0,0};
    for (int k0 = 0; k0 < HID; k0 += 4) {
        int ka = k0 + 2 * half;
        v2f a = *(const v2f*)(hr + ka);
        v2f b; b.x = W2[(size_t)ka * FOUT + lo]; b.y = W2[(size_t)(ka + 1) * FOUT + lo];
        c = wmma_k4(a, b, c);
    }
    for (int j = 0; j < 8; ++j) {
        int r = rowBase + j + 8 * half;
        out[(size_t)r * FOUT + lo] = c[j] + b2[lo];
    }
}

// ---------------------------------------------------------------------------
extern "C" void kernel_launch(void* const* d_in, const int* in_sizes, int n_in,
                              void* d_out, int out_size, void* d_ws, size_t ws_size,
                              hipStream_t stream) {
    const float* x     = (const float*)d_in[0];  // [B, M, 32]
    const float* Adj_t = (const float*)d_in[1];  // [64,64]
    const float* Adj_s = (const float*)d_in[2];  // [128,128]
    const float* s     = (const float*)d_in[3];  // [4]
    const float* H     = (const float*)d_in[4];  // [L,K,32,32]
    const float* W1    = (const float*)d_in[5];  // [32,32]
    const float* b1    = (const float*)d_in[6];  // [32]
    const float* W2    = (const float*)d_in[7];  // [32,16]
    const float* b2    = (const float*)d_in[8];  // [16]
    float* out = (float*)d_out;

    // Workspace layout (floats): Heff | hA | hB | u | v  (~8.4 MB total)
    float* ws   = (float*)d_ws;
    const size_t HSZ = (size_t)TB * TM * HID;    // 524288
    float* Heff = ws;                            // NLAYERS*32*32 = 2048
    float* hA   = ws + 2048;
    float* hB   = hA + HSZ;
    float* ub   = hB + HSZ;
    float* vb   = ub + HSZ;

    heff_kernel<<<(NLAYERS * HID * HID + 255) / 256, 256, 0, stream>>>(H, Heff);
    inproj_kernel<<<(TB * TM / 16) / 8, 256, 0, stream>>>(x, W1, b1, hA);

    float* hcur = hA;
    float* hnxt = hB;
    for (int l = 0; l < NLAYERS; ++l) {
        spatial_kernel<<<TB * TT, 512, 0, stream>>>(Adj_s, hcur, s, ub, vb);
        temporal_kernel<<<TB * TN, 256, 0, stream>>>(Adj_t, ub, vb,
                                                     Heff + (size_t)l * HID * HID, hnxt);
        float* tmp = hcur; hcur = hnxt; hnxt = tmp;
    }
    outproj_kernel<<<(TB * TM / 16) / 8, 256, 0, stream>>>(hcur, W2, b2, out);
}